// NMS_84825604096355
// MI455X (gfx1250) — compile-verified
//
#include <hip/hip_runtime.h>
#include <hip/hip_bf16.h>

// ---------------------------------------------------------------------------
// NMS for MI455X (gfx1250, wave32).
//   K1 hist      : 8192-bin LDS-privatized histogram of scores
//   K2 thresh    : single-WG suffix scan -> threshold bin T (>=1000 above T)
//   K3 compact   : emit (score_bits<<32)|(~index) keys for bins >= T
//   K4 sortgather: 1 WG bitonic sort (4096, desc) in LDS, gather top-1000 boxes
//   K5 suppress  : 63x63 tiles, 1 wave/tile; V_WMMA_F32_16X16X4_F32 computes
//                  area_i+area_j outer sum (exact); ballot packs bits
//   K6 reduce    : single wave sequential keep-propagation (no barriers)
//   K7 final     : out = boxes5 * keep
// ---------------------------------------------------------------------------

typedef __attribute__((ext_vector_type(2))) float v2f;
typedef __attribute__((ext_vector_type(8))) float v8f;

#define NBINS   8192
#define TOPK    1000
#define NPAD    1008            // 63 * 16
#define NTILES  63
#define CANDCAP 4096
#define SUPW    64              // u32 words per suppression row (63 used)

__device__ __forceinline__ int score_bin(float sc) {
    int b = (int)(sc * (float)NBINS);
    return b < 0 ? 0 : (b > NBINS - 1 ? NBINS - 1 : b);
}

// ---- K1: histogram ---------------------------------------------------------
__global__ void nms_hist(const float* __restrict__ x, int n, unsigned int* __restrict__ hist) {
    __shared__ unsigned int h[NBINS];
    for (int b = threadIdx.x; b < NBINS; b += blockDim.x) h[b] = 0u;
    __syncthreads();
    int i = blockIdx.x * blockDim.x + threadIdx.x;
    int stride = gridDim.x * blockDim.x;
    for (; i < n; i += stride) {
        float sc = x[(size_t)i * 5 + 4];
        atomicAdd(&h[score_bin(sc)], 1u);
    }
    __syncthreads();
    for (int b = threadIdx.x; b < NBINS; b += blockDim.x)
        if (h[b]) atomicAdd(&hist[b], h[b]);
}

// ---- K2: find threshold bin T ---------------------------------------------
__global__ void nms_thresh(const unsigned int* __restrict__ hist, unsigned int* __restrict__ Tout) {
    __shared__ unsigned int s[1024];
    int t = threadIdx.x;                         // 1024 threads, 8 bins each (descending)
    int hi = NBINS - 1 - t * 8;
    unsigned int sum = 0;
    #pragma unroll
    for (int b = 0; b < 8; ++b) sum += hist[hi - b];
    s[t] = sum;
    __syncthreads();
    for (int off = 1; off < 1024; off <<= 1) {   // inclusive scan (counts from top)
        unsigned int v = (t >= off) ? s[t - off] : 0u;
        __syncthreads();
        s[t] += v;
        __syncthreads();
    }
    unsigned int incl = s[t];
    unsigned int excl = incl - sum;
    if (t == 0 && s[1023] < TOPK) *Tout = 0u;    // degenerate fallback
    if (excl < TOPK && incl >= TOPK) {
        unsigned int c = excl, T = 0u;
        #pragma unroll
        for (int b = 0; b < 8; ++b) {
            c += hist[hi - b];
            if (c >= TOPK) { T = (unsigned int)(hi - b); break; }
        }
        *Tout = T;
    }
}

// ---- K3: compact candidates -----------------------------------------------
__global__ void nms_compact(const float* __restrict__ x, int n,
                            const unsigned int* __restrict__ Tp,
                            unsigned int* __restrict__ counter,
                            unsigned long long* __restrict__ cand) {
    unsigned int T = *Tp;
    int i = blockIdx.x * blockDim.x + threadIdx.x;
    int stride = gridDim.x * blockDim.x;
    for (; i < n; i += stride) {
        float sc = x[(size_t)i * 5 + 4];
        if ((unsigned int)score_bin(sc) >= T) {
            unsigned int pos = atomicAdd(counter, 1u);
            if (pos < CANDCAP) {
                unsigned int sb = __float_as_uint(sc);
                // key: score desc, then index asc (matches jax.lax.top_k ties)
                cand[pos] = ((unsigned long long)sb << 32) |
                            (unsigned long long)(0xFFFFFFFFu - (unsigned int)i);
            }
        }
    }
}

// ---- K4: bitonic sort + gather --------------------------------------------
__global__ void nms_sortgather(const unsigned long long* __restrict__ cand,
                               const unsigned int* __restrict__ counterp,
                               const float* __restrict__ x,
                               float4* __restrict__ boxes4,
                               float* __restrict__ scoreT,
                               float* __restrict__ areaT) {
    __shared__ unsigned long long k[CANDCAP];    // 32 KB
    int tid = threadIdx.x;                       // 512 threads
    unsigned int cnt = *counterp;
    if (cnt > CANDCAP) cnt = CANDCAP;
    for (int i = tid; i < CANDCAP; i += 512) k[i] = (i < (int)cnt) ? cand[i] : 0ull;
    __syncthreads();
    for (int kk = 2; kk <= CANDCAP; kk <<= 1) {
        for (int j = kk >> 1; j > 0; j >>= 1) {
            for (int i = tid; i < CANDCAP; i += 512) {
                int ixj = i ^ j;
                if (ixj > i) {
                    unsigned long long a = k[i], b = k[ixj];
                    bool descSeg = ((i & kk) == 0);
                    if (descSeg ? (a < b) : (a > b)) { k[i] = b; k[ixj] = a; }
                }
            }
            __syncthreads();
        }
    }
    for (int t = tid; t < NPAD; t += 512) {
        if (t < TOPK) {
            unsigned long long key = k[t];
            unsigned int idx = 0xFFFFFFFFu - (unsigned int)(key & 0xFFFFFFFFull);
            const float* row = x + (size_t)idx * 5;
            float x1 = row[0], y1 = row[1], x2 = row[2], y2 = row[3];
            boxes4[t] = make_float4(x1, y1, x2, y2);
            scoreT[t] = row[4];
            areaT[t]  = fmaxf(x2 - x1, 0.0f) * fmaxf(y2 - y1, 0.0f);
        } else {
            boxes4[t] = make_float4(0.f, 0.f, 0.f, 0.f);
            scoreT[t] = 0.f;
            areaT[t]  = 0.f;
        }
    }
}

// ---- K5: suppression bitmask via WMMA -------------------------------------
// One wave per 16x16 tile. D = A*B + C with:
//   A[m][*] = one 1 per row (rest 0)  ->  picks one (identical) row of B
//   B[k][n] = area_j[n] for all k     ->  layout-robust broadcast
//   C[m][n] = area_i[m]
// => D[m][n] = area_i[m] + area_j[n]  (exact, single RNE add)
__global__ void nms_suppress(const float4* __restrict__ boxes4,
                             const float* __restrict__ areaT,
                             unsigned int* __restrict__ sup) {
    int tj = blockIdx.x, ti = blockIdx.y;
    int lane = threadIdx.x;                      // 0..31, one wave
    int i0 = ti * 16, j0 = tj * 16;
    int n = lane & 15;
    int laneHi = lane >> 4;

    v2f a; a.x = (laneHi == 0) ? 1.0f : 0.0f; a.y = 0.0f;
    float aj = areaT[j0 + n];
    v2f b; b.x = aj; b.y = aj;
    v8f c;
    #pragma unroll
    for (int r = 0; r < 8; ++r) c[r] = areaT[i0 + r + laneHi * 8];

    v8f d = __builtin_amdgcn_wmma_f32_16x16x4_f32(
        /*neg_a=*/false, a, /*neg_b=*/false, b,
        /*c_mod=*/(short)0, c, /*reuse_a=*/false, /*reuse_b=*/false);

    float4 cb = boxes4[j0 + n];
    int gj = j0 + n;
    #pragma unroll
    for (int r = 0; r < 8; ++r) {
        int m = r + laneHi * 8;
        int gi = i0 + m;
        float4 rb = boxes4[i0 + m];
        float ix1 = fmaxf(rb.x, cb.x), iy1 = fmaxf(rb.y, cb.y);
        float ix2 = fminf(rb.z, cb.z), iy2 = fminf(rb.w, cb.w);
        float inter = fmaxf(ix2 - ix1, 0.0f) * fmaxf(iy2 - iy1, 0.0f);
        float uni = d[r] - inter;                // (a_i + a_j) - inter, as reference
        float iou = inter / (uni + 1e-9f);
        bool pred = (iou > 0.5f) && (gj > gi);
        unsigned int bal = (unsigned int)__ballot(pred);
        // C layout: bits 0-15 = row (i0+r), bits 16-31 = row (i0+8+r)
        if (lane == 0) {
            sup[(unsigned)(i0 + r)     * SUPW + (unsigned)tj] = bal & 0xFFFFu;
            sup[(unsigned)(i0 + 8 + r) * SUPW + (unsigned)tj] = bal >> 16;
        }
    }
}

// ---- K6: sequential keep propagation (single wave, no barriers) -----------
__global__ void nms_reduce(const unsigned int* __restrict__ sup,
                           unsigned int* __restrict__ removed) {
    int l = threadIdx.x;                         // 32 lanes; lane l owns words l and l+32
    unsigned int w0 = 0u, w1 = 0u;
    for (int i = 0; i < TOPK; ++i) {
        int wrd = i >> 4, bit = i & 15;
        unsigned int rw = (wrd < 32) ? __shfl(w0, wrd, 32) : __shfl(w1, wrd - 32, 32);
        if (((rw >> bit) & 1u) == 0u) {          // box i kept -> it suppresses
            w0 |= sup[(unsigned)i * SUPW + (unsigned)l];
            if (l < 31) w1 |= sup[(unsigned)i * SUPW + 32u + (unsigned)l];
        }
    }
    removed[l] = w0;
    removed[32 + l] = (l < 31) ? w1 : 0u;
}

// ---- K7: masked output -----------------------------------------------------
__global__ void nms_final(const float4* __restrict__ boxes4,
                          const float* __restrict__ scoreT,
                          const unsigned int* __restrict__ removed,
                          float* __restrict__ out) {
    int t = blockIdx.x * blockDim.x + threadIdx.x;
    if (t >= TOPK * 5) return;
    int i = t / 5, cidx = t % 5;
    float keep = (((removed[i >> 4] >> (i & 15)) & 1u) == 0u) ? 1.0f : 0.0f;
    float4 bx = boxes4[i];
    float v = (cidx == 0) ? bx.x : (cidx == 1) ? bx.y : (cidx == 2) ? bx.z
            : (cidx == 3) ? bx.w : scoreT[i];
    out[t] = v * keep;                           // multiply (preserve -0 semantics)
}

// ---- workspace layout (bytes) ---------------------------------------------
//      0 : hist     8192 * u32   = 32768
//  32768 : counter  u32 (memset together with hist)
//  33280 : cand     4096 * u64   = 32768
//  66048 : boxes4   1008 * float4= 16128
//  82176 : scoreT   1008 * f32   =  4032
//  86208 : areaT    1008 * f32   =  4032
//  90240 : sup      1008*64*u32  = 258048
// 348288 : removed  64 * u32     =   256   (total ~349 KB)
// Tp at 33024.

extern "C" void kernel_launch(void* const* d_in, const int* in_sizes, int n_in,
                              void* d_out, int out_size, void* d_ws, size_t ws_size,
                              hipStream_t stream) {
    const float* x = (const float*)d_in[0];
    int n = in_sizes[0] / 5;

    char* ws = (char*)d_ws;
    unsigned int*       hist    = (unsigned int*)(ws + 0);
    unsigned int*       counter = (unsigned int*)(ws + 32768);
    unsigned int*       Tp      = (unsigned int*)(ws + 33024);
    unsigned long long* cand    = (unsigned long long*)(ws + 33280);
    float4*             boxes4  = (float4*)(ws + 66048);
    float*              scoreT  = (float*)(ws + 82176);
    float*              areaT   = (float*)(ws + 86208);
    unsigned int*       sup     = (unsigned int*)(ws + 90240);
    unsigned int*       removed = (unsigned int*)(ws + 348288);

    // zero histogram + candidate counter every launch (graph-capture safe)
    hipMemsetAsync(ws, 0, 32768 + 256, stream);

    nms_hist   <<<256, 256, 0, stream>>>(x, n, hist);
    nms_thresh <<<1, 1024, 0, stream>>>(hist, Tp);
    nms_compact<<<256, 256, 0, stream>>>(x, n, Tp, counter, cand);
    nms_sortgather<<<1, 512, 0, stream>>>(cand, counter, x, boxes4, scoreT, areaT);
    dim3 tiles(NTILES, NTILES);
    nms_suppress<<<tiles, 32, 0, stream>>>(boxes4, areaT, sup);
    nms_reduce <<<1, 32, 0, stream>>>(sup, removed);
    nms_final  <<<(TOPK * 5 + 255) / 256, 256, 0, stream>>>(boxes4, scoreT, removed, (float*)d_out);
}